// Net_GCN_mnist_85229331021943
// MI455X (gfx1250) — compile-verified
//
#include <hip/hip_runtime.h>

// ---------------------------------------------------------------------------
// MI455X (gfx1250) implementation of the GCN pipeline.
// All heavy GEMMs run on V_WMMA_F32_16X16X4_F32 (fp32 in / fp32 acc).
// ---------------------------------------------------------------------------

typedef __attribute__((ext_vector_type(2))) float v2f;
typedef __attribute__((ext_vector_type(8))) float v8f;

#define WMMA_F32X4(a, b, c) \
  __builtin_amdgcn_wmma_f32_16x16x4_f32(false, (a), false, (b), (short)0, (c), false, false)

// ------------------------------- constants ---------------------------------
#define BATCH   32
#define C_IN    3
#define HW      64
#define CL1_F   64
#define V1      1024
#define CL2_F   128
#define CHEB_K  16
#define V2      256
#define CL3_F   256
#define FC1_F   512
#define FC1_IN  16384   // 256 * 64
#define FC2_F   10

// ===========================================================================
// Kernel 1: conv5x5 + bias + relu + maxpool2x2 + gather(node_index) + perm
// writes x0 in graph-conv layout: x0[v * 2048 + c * 32 + b]
// ===========================================================================
__global__ __launch_bounds__(256) void conv_pool_gather_kernel(
    const float* __restrict__ x, const float* __restrict__ w,
    const float* __restrict__ bias, const int* __restrict__ node_index,
    const int* __restrict__ perm, float* __restrict__ x0) {
  int t = blockIdx.x * 256 + threadIdx.x;   // 1024 * 64 * 32 = 2,097,152
  int b = t & 31;
  int c = (t >> 5) & 63;
  int p = t >> 11;                          // node index 0..1023

  int vsrc = perm[p];
  int gi = node_index[2 * vsrc + 0];
  int gj = node_index[2 * vsrc + 1];

  const float* wc = w + c * (C_IN * 25);
  const float* xb = x + b * (C_IN * HW * HW);
  float bval = bias[c];
  float best = -3.0e38f;

#pragma unroll
  for (int di = 0; di < 2; ++di) {
#pragma unroll
    for (int dj = 0; dj < 2; ++dj) {
      int h = gi * 2 + di;
      int wp = gj * 2 + dj;
      float s = bval;
#pragma unroll
      for (int ci = 0; ci < C_IN; ++ci) {
        const float* xc = xb + ci * (HW * HW);
        const float* wk = wc + ci * 25;
#pragma unroll
        for (int kh = 0; kh < 5; ++kh) {
          int hh = h + kh - 2;
          if ((unsigned)hh < (unsigned)HW) {
#pragma unroll
            for (int kw = 0; kw < 5; ++kw) {
              int ww = wp + kw - 2;
              if ((unsigned)ww < (unsigned)HW)
                s += xc[hh * HW + ww] * wk[kh * 5 + kw];
            }
          }
        }
      }
      best = fmaxf(best, s);
    }
  }
  best = fmaxf(best, 0.0f);
  x0[p * (CL1_F * BATCH) + c * BATCH + b] = best;
}

// ===========================================================================
// BatchNorm over (B, V) per channel: stats then normalize
// ===========================================================================
__global__ __launch_bounds__(256) void bn_stats_kernel(
    const float* __restrict__ x0, float* __restrict__ stats) {
  __shared__ float s1[256];
  __shared__ float s2[256];
  int c = blockIdx.x;                       // 0..63
  int tid = threadIdx.x;
  float sum = 0.0f, sq = 0.0f;
  for (int i = tid; i < BATCH * V1; i += 256) {
    int v = i >> 5, b = i & 31;
    float val = x0[v * (CL1_F * BATCH) + c * BATCH + b];
    sum += val;
    sq += val * val;
  }
  s1[tid] = sum; s2[tid] = sq;
  __syncthreads();
  for (int off = 128; off > 0; off >>= 1) {
    if (tid < off) { s1[tid] += s1[tid + off]; s2[tid] += s2[tid + off]; }
    __syncthreads();
  }
  if (tid == 0) {
    float inv_n = 1.0f / (float)(BATCH * V1);
    float m = s1[0] * inv_n;
    float var = s2[0] * inv_n - m * m;
    stats[c * 2 + 0] = m;
    stats[c * 2 + 1] = rsqrtf(var + 1e-5f);
  }
}

__global__ __launch_bounds__(256) void bn_norm_kernel(
    float* __restrict__ x0, const float* __restrict__ stats) {
  int t = blockIdx.x * 256 + threadIdx.x;   // 2,097,152
  int c = (t >> 5) & 63;
  x0[t] = (x0[t] - stats[c * 2 + 0]) * stats[c * 2 + 1];
}

// ===========================================================================
// Initialize a (M x F) accumulator with per-column bias (F = power of 2)
// ===========================================================================
__global__ __launch_bounds__(256) void bias_init_kernel(
    float* __restrict__ out, const float* __restrict__ bias, int fmask) {
  int t = blockIdx.x * 256 + threadIdx.x;
  out[t] = bias[t & fmask];
}

// ===========================================================================
// Chebyshev recurrence GEMM: Out = alpha * (L @ X) + beta * Prev
// L: (M x K) row-major, X: (K x N) row-major, Out/Prev: (M x N).
// WG = 8 waves; each wave computes a 16x64 tile (4 WMMA accumulators,
// A-fragment reused 4x). Grid: (M/128, N/64). No divergence (EXEC all-1s).
// ===========================================================================
__global__ __launch_bounds__(256) void cheb_gemm_kernel(
    const float* __restrict__ L, const float* __restrict__ X,
    const float* __restrict__ Prev, float* __restrict__ Out,
    int M, int N, int K, float alpha, float beta) {
  int lane = threadIdx.x & 31;
  int wave = threadIdx.x >> 5;
  int m0 = (blockIdx.x * 8 + wave) * 16;
  int n0 = blockIdx.y * 64;

  int krow = (lane >> 4) << 1;                         // 0 or 2
  const float* Arow = L + (m0 + (lane & 15)) * K + krow;
  const float* Xcol = X + n0 + (lane & 15);

  v8f acc0 = {}, acc1 = {}, acc2 = {}, acc3 = {};

#pragma unroll 2
  for (int k0 = 0; k0 < K; k0 += 4) {
    v2f a = {Arow[k0], Arow[k0 + 1]};
    const float* xp = Xcol + (k0 + krow) * N;
    v2f b0 = {xp[0],  xp[N + 0]};
    v2f b1 = {xp[16], xp[N + 16]};
    v2f b2 = {xp[32], xp[N + 32]};
    v2f b3 = {xp[48], xp[N + 48]};
    acc0 = WMMA_F32X4(a, b0, acc0);
    acc1 = WMMA_F32X4(a, b1, acc1);
    acc2 = WMMA_F32X4(a, b2, acc2);
    acc3 = WMMA_F32X4(a, b3, acc3);
  }

  int rbase = m0 + ((lane >> 4) << 3);
  int col = n0 + (lane & 15);
  bool hp = (Prev != nullptr);
#pragma unroll
  for (int r = 0; r < 8; ++r) {
    int row = rbase + r;
    float v0 = alpha * acc0[r];
    float v1 = alpha * acc1[r];
    float v2 = alpha * acc2[r];
    float v3 = alpha * acc3[r];
    if (hp) {
      const float* pp = Prev + row * N + col;
      v0 += beta * pp[0];
      v1 += beta * pp[16];
      v2 += beta * pp[32];
      v3 += beta * pp[48];
    }
    float* op = Out + row * N + col;
    op[0]  = v0;
    op[16] = v1;
    op[32] = v2;
    op[48] = v3;
  }
}

// ===========================================================================
// Per-Chebyshev-order accumulating projection GEMM:
//   Out[b*V + v, f] += sum_c T[v, c*32 + b] * W[f, c*Kcheb + kidx]
// T: (V x C*32) row-major, W: (F x C*Kcheb) row-major, Out: (32V x F).
// Wave tile 16x64, WG = 8 waves (128 rows). Grid: (32V/128, F/64).
// ===========================================================================
__global__ __launch_bounds__(256) void cheb_acc_gemm_kernel(
    const float* __restrict__ T, const float* __restrict__ W,
    float* __restrict__ Out, int Vshift, int C, int F, int Kcheb, int kidx) {
  int lane = threadIdx.x & 31;
  int wave = threadIdx.x >> 5;
  int m0 = (blockIdx.x * 8 + wave) * 16;
  int n0 = blockIdx.y * 64;

  int Cs = C << 5;                       // T row stride (C * 32)
  int CK = C * Kcheb;
  int Vmask = (1 << Vshift) - 1;

  int arow = m0 + (lane & 15);
  int av = arow & Vmask;
  int ab = arow >> Vshift;
  const float* Trow = T + av * Cs + ab;  // index with + c*32

  int klane2 = (lane >> 4) << 1;         // 0 or 2
  int fcol = n0 + (lane & 15);
  const float* W0 = W + (fcol +  0) * CK + kidx;
  const float* W1 = W + (fcol + 16) * CK + kidx;
  const float* W2 = W + (fcol + 32) * CK + kidx;
  const float* W3 = W + (fcol + 48) * CK + kidx;

  v8f acc0 = {}, acc1 = {}, acc2 = {}, acc3 = {};

#pragma unroll 2
  for (int c0 = 0; c0 < C; c0 += 4) {
    int c = c0 + klane2;
    v2f a  = {Trow[c << 5], Trow[(c + 1) << 5]};
    v2f b0 = {W0[c * Kcheb], W0[(c + 1) * Kcheb]};
    v2f b1 = {W1[c * Kcheb], W1[(c + 1) * Kcheb]};
    v2f b2 = {W2[c * Kcheb], W2[(c + 1) * Kcheb]};
    v2f b3 = {W3[c * Kcheb], W3[(c + 1) * Kcheb]};
    acc0 = WMMA_F32X4(a, b0, acc0);
    acc1 = WMMA_F32X4(a, b1, acc1);
    acc2 = WMMA_F32X4(a, b2, acc2);
    acc3 = WMMA_F32X4(a, b3, acc3);
  }

  int rbase = m0 + ((lane >> 4) << 3);
  int col = n0 + (lane & 15);
#pragma unroll
  for (int r = 0; r < 8; ++r) {
    float* o = Out + (rbase + r) * F + col;
    o[0]  += acc0[r];
    o[16] += acc1[r];
    o[32] += acc2[r];
    o[48] += acc3[r];
  }
}

// ===========================================================================
// relu + pool-by-4 along V, repack into next stage's graph layout
// stage 1: out1(32768 x 128) -> x0_2[v2*4096 + c*32 + b]
// ===========================================================================
__global__ __launch_bounds__(256) void pool1_kernel(
    const float* __restrict__ out1, float* __restrict__ x0_2) {
  int t = blockIdx.x * 256 + threadIdx.x;   // 256 * 128 * 32 = 1,048,576
  int b = t & 31;
  int c = (t >> 5) & 127;
  int v2 = t >> 12;
  const float* base = out1 + (b * V1 + v2 * 4) * CL2_F + c;
  float m = base[0];
  m = fmaxf(m, base[CL2_F]);
  m = fmaxf(m, base[2 * CL2_F]);
  m = fmaxf(m, base[3 * CL2_F]);
  x0_2[t] = fmaxf(m, 0.0f);
}

// stage 2: out2(8192 x 256) -> fc_in[b*16384 + c*64 + v3]
__global__ __launch_bounds__(256) void pool2_kernel(
    const float* __restrict__ out2, float* __restrict__ fc_in) {
  int t = blockIdx.x * 256 + threadIdx.x;   // 32 * 256 * 64 = 524,288
  int v3 = t & 63;
  int c = (t >> 6) & 255;
  int b = t >> 14;
  const float* base = out2 + (b * V2 + v3 * 4) * CL3_F + c;
  float m = base[0];
  m = fmaxf(m, base[CL3_F]);
  m = fmaxf(m, base[2 * CL3_F]);
  m = fmaxf(m, base[3 * CL3_F]);
  fc_in[t] = fmaxf(m, 0.0f);
}

// ===========================================================================
// FC1 via WMMA: out(32 x 512) = relu(fc_in(32 x 16384) @ W^T + bias)
// WG = 8 waves: (m-tile = wave&1) x (64-wide n-group = wave>>1). Grid: 2.
// ===========================================================================
__global__ __launch_bounds__(256) void fc1_kernel(
    const float* __restrict__ in, const float* __restrict__ W,
    const float* __restrict__ bias, float* __restrict__ out) {
  int lane = threadIdx.x & 31;
  int wave = threadIdx.x >> 5;
  int m0 = (wave & 1) * 16;
  int n0 = blockIdx.x * 256 + (wave >> 1) * 64;
  const int K = FC1_IN;

  int klane2 = (lane >> 4) << 1;
  const float* Arow = in + (m0 + (lane & 15)) * K + klane2;
  int fcol = n0 + (lane & 15);
  const float* W0 = W + (fcol +  0) * K + klane2;
  const float* W1 = W + (fcol + 16) * K + klane2;
  const float* W2 = W + (fcol + 32) * K + klane2;
  const float* W3 = W + (fcol + 48) * K + klane2;

  v8f acc0 = {}, acc1 = {}, acc2 = {}, acc3 = {};

#pragma unroll 4
  for (int k0 = 0; k0 < K; k0 += 4) {
    v2f a  = {Arow[k0], Arow[k0 + 1]};
    v2f b0 = {W0[k0], W0[k0 + 1]};
    v2f b1 = {W1[k0], W1[k0 + 1]};
    v2f b2 = {W2[k0], W2[k0 + 1]};
    v2f b3 = {W3[k0], W3[k0 + 1]};
    acc0 = WMMA_F32X4(a, b0, acc0);
    acc1 = WMMA_F32X4(a, b1, acc1);
    acc2 = WMMA_F32X4(a, b2, acc2);
    acc3 = WMMA_F32X4(a, b3, acc3);
  }

  int rbase = m0 + ((lane >> 4) << 3);
  int col = n0 + (lane & 15);
#pragma unroll
  for (int r = 0; r < 8; ++r) {
    float* o = out + (rbase + r) * FC1_F + col;
    o[0]  = fmaxf(acc0[r] + bias[col +  0], 0.0f);
    o[16] = fmaxf(acc1[r] + bias[col + 16], 0.0f);
    o[32] = fmaxf(acc2[r] + bias[col + 32], 0.0f);
    o[48] = fmaxf(acc3[r] + bias[col + 48], 0.0f);
  }
}

// ===========================================================================
// FC2 (32x512)@(512x10) + bias  — tiny, plain VALU
// ===========================================================================
__global__ __launch_bounds__(256) void fc2_kernel(
    const float* __restrict__ in, const float* __restrict__ W,
    const float* __restrict__ bias, float* __restrict__ out) {
  int t = blockIdx.x * 256 + threadIdx.x;
  if (t >= BATCH * FC2_F) return;
  int b = t / FC2_F;
  int f = t % FC2_F;
  float s = bias[f];
  const float* ib = in + b * FC1_F;
  const float* wf = W + f * FC1_F;
  for (int k = 0; k < FC1_F; ++k) s += ib[k] * wf[k];
  out[t] = s;
}

// ===========================================================================
// Host orchestration
// ===========================================================================
extern "C" void kernel_launch(void* const* d_in, const int* in_sizes, int n_in,
                              void* d_out, int out_size, void* d_ws, size_t ws_size,
                              hipStream_t stream) {
  (void)in_sizes; (void)n_in; (void)out_size; (void)ws_size;

  const float* x          = (const float*)d_in[0];
  const float* conv1_w    = (const float*)d_in[1];
  const float* conv1_b    = (const float*)d_in[2];
  const float* L1         = (const float*)d_in[3];
  const float* L2         = (const float*)d_in[4];
  const int*   node_index = (const int*)d_in[5];
  const int*   perm       = (const int*)d_in[6];
  const float* k2_w       = (const float*)d_in[7];
  const float* k2_b       = (const float*)d_in[8];
  const float* k3_w       = (const float*)d_in[9];
  const float* k3_b       = (const float*)d_in[10];
  const float* fc1_w      = (const float*)d_in[11];
  const float* fc1_b      = (const float*)d_in[12];
  const float* fc2_w      = (const float*)d_in[13];
  const float* fc2_b      = (const float*)d_in[14];

  // Workspace carve-up (floats). Total ~16.27M floats ≈ 65 MB (L2-resident).
  float* ws = (float*)d_ws;
  const size_t T1_SZ = (size_t)V1 * CL1_F * BATCH;   // 2,097,152
  const size_t O1_SZ = (size_t)BATCH * V1 * CL2_F;   // 4,194,304
  const size_t T2_SZ = (size_t)V2 * CL2_F * BATCH;   // 1,048,576
  const size_t O2_SZ = (size_t)BATCH * V2 * CL3_F;   // 2,097,152
  float* t1[3] = {ws, ws + T1_SZ, ws + 2 * T1_SZ};
  float* out1  = ws + 3 * T1_SZ;
  float* t2[3] = {out1 + O1_SZ, out1 + O1_SZ + T2_SZ, out1 + O1_SZ + 2 * T2_SZ};
  float* out2  = t2[0] + 3 * T2_SZ;
  float* fc_in = out2 + O2_SZ;
  float* fc1o  = fc_in + (size_t)BATCH * FC1_IN;
  float* stats = fc1o + (size_t)BATCH * FC1_F;

  // --- front end: conv + relu + pool + gather + perm -> x0 (= T_0) ---
  conv_pool_gather_kernel<<<8192, 256, 0, stream>>>(x, conv1_w, conv1_b,
                                                    node_index, perm, t1[0]);
  bn_stats_kernel<<<CL1_F, 256, 0, stream>>>(t1[0], stats);
  bn_norm_kernel<<<8192, 256, 0, stream>>>(t1[0], stats);

  // --- stage 1 Chebyshev graph conv (V=1024, C=64 -> F=128, K=16) ---
  bias_init_kernel<<<O1_SZ / 256, 256, 0, stream>>>(out1, k2_b, CL2_F - 1);
  dim3 g1(V1 / 128, (CL1_F * BATCH) / 64);             // (8, 32)
  dim3 ga1((BATCH * V1) / 128, CL2_F / 64);            // (256, 2)
  cheb_acc_gemm_kernel<<<ga1, 256, 0, stream>>>(t1[0], k2_w, out1, 10, CL1_F,
                                                CL2_F, CHEB_K, 0);
  cheb_gemm_kernel<<<g1, 256, 0, stream>>>(L1, t1[0], nullptr, t1[1], V1,
                                           CL1_F * BATCH, V1, 1.0f, 0.0f);
  cheb_acc_gemm_kernel<<<ga1, 256, 0, stream>>>(t1[1], k2_w, out1, 10, CL1_F,
                                                CL2_F, CHEB_K, 1);
  for (int k = 2; k < CHEB_K; ++k) {
    float* Tn  = t1[k % 3];
    float* Tm1 = t1[(k - 1) % 3];
    float* Tm2 = t1[(k - 2) % 3];
    cheb_gemm_kernel<<<g1, 256, 0, stream>>>(L1, Tm1, Tm2, Tn, V1,
                                             CL1_F * BATCH, V1, 2.0f, -1.0f);
    cheb_acc_gemm_kernel<<<ga1, 256, 0, stream>>>(Tn, k2_w, out1, 10, CL1_F,
                                                  CL2_F, CHEB_K, k);
  }
  pool1_kernel<<<4096, 256, 0, stream>>>(out1, t2[0]);

  // --- stage 2 Chebyshev graph conv (V=256, C=128 -> F=256, K=16) ---
  bias_init_kernel<<<O2_SZ / 256, 256, 0, stream>>>(out2, k3_b, CL3_F - 1);
  dim3 g2(V2 / 128, (CL2_F * BATCH) / 64);             // (2, 64)
  dim3 ga2((BATCH * V2) / 128, CL3_F / 64);            // (64, 4)
  cheb_acc_gemm_kernel<<<ga2, 256, 0, stream>>>(t2[0], k3_w, out2, 8, CL2_F,
                                                CL3_F, CHEB_K, 0);
  cheb_gemm_kernel<<<g2, 256, 0, stream>>>(L2, t2[0], nullptr, t2[1], V2,
                                           CL2_F * BATCH, V2, 1.0f, 0.0f);
  cheb_acc_gemm_kernel<<<ga2, 256, 0, stream>>>(t2[1], k3_w, out2, 8, CL2_F,
                                                CL3_F, CHEB_K, 1);
  for (int k = 2; k < CHEB_K; ++k) {
    float* Tn  = t2[k % 3];
    float* Tm1 = t2[(k - 1) % 3];
    float* Tm2 = t2[(k - 2) % 3];
    cheb_gemm_kernel<<<g2, 256, 0, stream>>>(L2, Tm1, Tm2, Tn, V2,
                                             CL2_F * BATCH, V2, 2.0f, -1.0f);
    cheb_acc_gemm_kernel<<<ga2, 256, 0, stream>>>(Tn, k3_w, out2, 8, CL2_F,
                                                  CL3_F, CHEB_K, k);
  }
  pool2_kernel<<<2048, 256, 0, stream>>>(out2, fc_in);

  // --- classifier head ---
  fc1_kernel<<<2, 256, 0, stream>>>(fc_in, fc1_w, fc1_b, fc1o);
  fc2_kernel<<<2, 256, 0, stream>>>(fc1o, fc2_w, fc2_b, (float*)d_out);
}